// TFN_15135464751723
// MI455X (gfx1250) — compile-verified
//
#include <hip/hip_runtime.h>
#include <math.h>

typedef __attribute__((ext_vector_type(2))) float v2f;
typedef __attribute__((ext_vector_type(8))) float v8f;

#define N_NODES 20000
#define N_EDGES 640000

#define INV_SQRT32 0.17677669529663687f   // 1/sqrt(32)  (node linears, mul0)
#define INV_4      0.25f                  // 1/sqrt(16)  (node linears, mul1)
#define INV_SQRT10 0.31622776601683794f   // 1/sqrt(10)  (fc0, folded into B tiles)
#define INV_8      0.125f                 // 1/sqrt(64)  (fc1, folded into LDS stage)
#define INV_SQRT3  0.5773502691896258f    // m_d normalization
#define C_S        0.3826834323650898f    // sin(pi/8)
#define C_X        0.9238795325112867f    // cos(pi/8)
#define Z_SCALE    0.025515518153991442f  // 1/sqrt(32) * 1/sqrt(48) = 1/sqrt(1536)

// ---------------------------------------------------------------------------
// Kernel 1: per-node linears (self-connection s and lin1 y), attr is scalar.
// ---------------------------------------------------------------------------
__global__ void tfn_node_prep(const float* __restrict__ xin,
                              const float* __restrict__ attr,
                              const float* __restrict__ Wsc0,
                              const float* __restrict__ Wsc1,
                              const float* __restrict__ Wl10,
                              const float* __restrict__ Wl11,
                              float* __restrict__ y0, float* __restrict__ y1,
                              float* __restrict__ s0, float* __restrict__ s1) {
  int tid = blockIdx.x * blockDim.x + threadIdx.x;
  if (tid >= N_NODES * 80) return;
  int n = tid / 80;
  int j = tid - n * 80;
  float a = attr[n];
  if (j < 32) {
    float accy = 0.f, accs = 0.f;
    const float* x = xin + n * 80;
#pragma unroll
    for (int u = 0; u < 32; ++u) {
      float xv = x[u];
      accy += xv * Wl10[u * 32 + j];
      accs += xv * Wsc0[u * 32 + j];
    }
    y0[n * 32 + j] = accy * a * INV_SQRT32;
    s0[n * 32 + j] = accs * a * INV_SQRT32;
  } else {
    int k = j - 32;
    int v = k / 3;
    int i = k - v * 3;
    float accy = 0.f, accs = 0.f;
    const float* x = xin + n * 80 + 32 + i;
#pragma unroll
    for (int u = 0; u < 16; ++u) {
      float xv = x[u * 3];
      accy += xv * Wl11[u * 16 + v];
      accs += xv * Wsc1[u * 16 + v];
    }
    y1[n * 48 + k] = accy * a * INV_4;
    s1[n * 48 + k] = accs * a * INV_4;
  }
}

// ---------------------------------------------------------------------------
// Kernel 2: per-edge MLP via f32 WMMA (16 edges/wave), tensor product,
// scatter-add to agg0/agg1 with f32 global atomics (L2-resident).
// f32 WMMA operand model (ISA 7.12.2):
//   A[m][k]: lane = (k>=2)*16 + m, vgpr = k&1   (v2f per lane)
//   B[k][n]: lane = (k>=2)*16 + n, vgpr = k&1   (v2f per lane)
//   C/D:     lane<16: M=r,N=lane ; lane>=16: M=8+r,N=lane-16   (v8f)
// ---------------------------------------------------------------------------
__global__ __launch_bounds__(128) void tfn_edge_kernel(
    const float* __restrict__ es,   // [E,10] edge_scalars
    const float* __restrict__ ea,   // [E,4]  edge_attr
    const int*   __restrict__ eidx, // [2,E]
    const float* __restrict__ Wf0,  // [10,64]
    const float* __restrict__ Wf1,  // [64,96]
    const float* __restrict__ y0g,  // [N,32]
    const float* __restrict__ y1g,  // [N,48]
    float* __restrict__ agg0,       // [N,48]
    float* __restrict__ agg1)       // [N,48,3]
{
  __shared__ __align__(16) float w1lds[6144];     // W_fc1 swizzled B tiles (24KB)
  __shared__ __align__(16) float buf[4][1536];    // per-wave scratch: h then w

  const int lane  = threadIdx.x & 31;
  const int wv    = threadIdx.x >> 5;
  const int nloc  = lane & 15;
  const int khalf = lane >> 4;

  // Stage W_fc1 into LDS pre-swizzled AND pre-scaled by 1/sqrt(64):
  // tile (kt,nt), lane l -> {W[k][n], W[k+1][n]} * 0.125
  for (int idx = threadIdx.x; idx < 3072; idx += 128) {
    int kt  = idx / 192;
    int rem = idx - kt * 192;
    int nt  = rem >> 5;
    int l   = rem & 31;
    int n   = nt * 16 + (l & 15);
    int k   = kt * 4 + (l >> 4) * 2;
    w1lds[idx * 2 + 0] = Wf1[k * 96 + n] * INV_8;
    w1lds[idx * 2 + 1] = Wf1[(k + 1) * 96 + n] * INV_8;
  }

  // Preload W_fc0 B tiles into registers, pre-scaled by 1/sqrt(10) (K pad 10->12)
  v2f bfc0[3][4];
#pragma unroll
  for (int ks = 0; ks < 3; ++ks) {
#pragma unroll
    for (int nt = 0; nt < 4; ++nt) {
      int k = ks * 4 + khalf * 2;
      int n = nt * 16 + nloc;
      v2f b;
      b.x = (k     < 10) ? Wf0[k * 64 + n]       * INV_SQRT10 : 0.f;
      b.y = (k + 1 < 10) ? Wf0[(k + 1) * 64 + n] * INV_SQRT10 : 0.f;
      bfc0[ks][nt] = b;
    }
  }
  __syncthreads();

  float* mybuf = buf[wv];
  const int totalWaves = gridDim.x * 4;
  const int wid = blockIdx.x * 4 + wv;

  for (int eb = wid; eb < N_EDGES / 16; eb += totalWaves) {
    const int e0i = eb * 16;

    // --- GEMM1 A tiles: 16 edges x K (edge_scalars), zero-padded past K=9
    v2f a1[3];
#pragma unroll
    for (int ks = 0; ks < 3; ++ks) {
      int k = ks * 4 + khalf * 2;
      v2f av; av.x = 0.f; av.y = 0.f;
      if (k <= 8) {
        const float* p = es + (size_t)(e0i + nloc) * 10 + k;
        av.x = p[0]; av.y = p[1];
      }
      a1[ks] = av;
    }

    // --- GEMM1: h = silu(es @ Wf0 / sqrt(10)); store h in LDS in A-layout
#pragma unroll
    for (int nt = 0; nt < 4; ++nt) {
      v8f c = {0.f, 0.f, 0.f, 0.f, 0.f, 0.f, 0.f, 0.f};
#pragma unroll
      for (int ks = 0; ks < 3; ++ks) {
        c = __builtin_amdgcn_wmma_f32_16x16x4_f32(
            false, a1[ks], false, bfc0[ks][nt], (short)0, c, false, false);
      }
      int kcol = nt * 16 + nloc;     // column of h == K index of GEMM2
      int kt2  = kcol >> 2;
      int kh2  = (kcol >> 1) & 1;
      int jj   = kcol & 1;
      int mbase = khalf * 8;
#pragma unroll
      for (int r = 0; r < 8; ++r) {
        float hv = c[r];             // already scaled via bfc0
        // silu(x) = x * sigmoid(x); v_exp + v_rcp (no IEEE divide)
        hv = hv * __builtin_amdgcn_rcpf(1.f + __expf(-hv));
        mybuf[(kt2 * 32 + kh2 * 16 + (mbase + r)) * 2 + jj] = hv;
      }
    }
    asm volatile("s_wait_dscnt 0" ::: "memory");

    // --- GEMM2: w[16x96] = h[16x64] @ (Wf1/8)[64x96]
    v8f wacc[6];
#pragma unroll
    for (int nt2 = 0; nt2 < 6; ++nt2) {
      v8f c = {0.f, 0.f, 0.f, 0.f, 0.f, 0.f, 0.f, 0.f};
#pragma unroll
      for (int kt = 0; kt < 16; ++kt) {
        v2f a2 = *(const v2f*)&mybuf[(kt * 32 + lane) * 2];
        v2f b2 = *(const v2f*)&w1lds[((kt * 6 + nt2) * 32 + lane) * 2];
        c = __builtin_amdgcn_wmma_f32_16x16x4_f32(
            false, a2, false, b2, (short)0, c, false, false);
      }
      wacc[nt2] = c;
    }
    asm volatile("s_wait_dscnt 0" ::: "memory");

    // store w row-major [16][96] into the same per-wave buffer
#pragma unroll
    for (int nt2 = 0; nt2 < 6; ++nt2) {
      int n = nt2 * 16 + nloc;
#pragma unroll
      for (int r = 0; r < 8; ++r) {
        mybuf[(khalf * 8 + r) * 96 + n] = wacc[nt2][r];
      }
    }
    asm volatile("s_wait_dscnt 0" ::: "memory");

    // --- tensor product + scatter-add (per edge, lanes over channels)
    for (int el = 0; el < 16; ++el) {
      int e   = e0i + el;
      int src = eidx[e];
      int dst = eidx[N_EDGES + e];
      float4 eav = *(const float4*)(ea + (size_t)e * 4);
      float ee0 = eav.x;
      float e1x = eav.y;
      float e1y = eav.z;
      float e1z = eav.w;
      const float* wrow = mybuf + el * 96;

      // channels 0..31: m_a (scalar path) and m_b (vector from scalar)
      {
        int ch   = lane;
        float x  = y0g[(size_t)src * 32 + ch];
        float wa = wrow[ch];
        float wb = wrow[32 + ch];
        float wbx = wb * x;
        atomicAdd(agg0 + (size_t)dst * 48 + ch, wa * x * ee0);
        float* a1p = agg1 + ((size_t)dst * 48 + ch) * 3;
        atomicAdd(a1p + 0, wbx * e1x);
        atomicAdd(a1p + 1, wbx * e1y);
        atomicAdd(a1p + 2, wbx * e1z);
      }
      // channels 32..47: m_c (vector from vector) and m_d (dot product)
      if (lane < 16) {
        int cch = lane;
        const float* vp = y1g + (size_t)src * 48 + cch * 3;
        float vx = vp[0], vy = vp[1], vz = vp[2];
        float wc = wrow[64 + cch];
        float wd = wrow[80 + cch];
        float dotv = vx * e1x + vy * e1y + vz * e1z;
        atomicAdd(agg0 + (size_t)dst * 48 + 32 + cch, wd * dotv * INV_SQRT3);
        float* a1p = agg1 + ((size_t)dst * 48 + 32 + cch) * 3;
        float wce0 = wc * ee0;
        atomicAdd(a1p + 0, wce0 * vx);
        atomicAdd(a1p + 1, wce0 * vy);
        atomicAdd(a1p + 2, wce0 * vz);
      }
    }
  }
}

// ---------------------------------------------------------------------------
// Kernel 3: final lin2 + equivariant mix: out = sin(pi/8)*s + cos(pi/8)*z
// ---------------------------------------------------------------------------
__global__ void tfn_node_out(const float* __restrict__ agg0,
                             const float* __restrict__ agg1,
                             const float* __restrict__ Wl20,
                             const float* __restrict__ Wl21,
                             const float* __restrict__ s0,
                             const float* __restrict__ s1,
                             float* __restrict__ out) {
  int tid = blockIdx.x * blockDim.x + threadIdx.x;
  if (tid >= N_NODES * 80) return;
  int n = tid / 80;
  int j = tid - n * 80;
  if (j < 32) {
    float z = 0.f;
    const float* ar = agg0 + (size_t)n * 48;
#pragma unroll
    for (int u = 0; u < 48; ++u) z += ar[u] * Wl20[u * 32 + j];
    out[tid] = C_S * s0[n * 32 + j] + C_X * z * Z_SCALE;
  } else {
    int k = j - 32;
    int v = k / 3;
    int i = k - v * 3;
    float z = 0.f;
    const float* ar = agg1 + (size_t)n * 144 + i;
#pragma unroll
    for (int u = 0; u < 48; ++u) z += ar[u * 3] * Wl21[u * 16 + v];
    out[tid] = C_S * s1[n * 48 + k] + C_X * z * Z_SCALE;
  }
}

// ---------------------------------------------------------------------------
extern "C" void kernel_launch(void* const* d_in, const int* in_sizes, int n_in,
                              void* d_out, int out_size, void* d_ws, size_t ws_size,
                              hipStream_t stream) {
  const float* node_input = (const float*)d_in[0];
  const float* node_attr  = (const float*)d_in[1];
  const float* edge_attr  = (const float*)d_in[2];
  const float* edge_scal  = (const float*)d_in[3];
  const float* W_sc0      = (const float*)d_in[4];
  const float* W_sc1      = (const float*)d_in[5];
  const float* W_l1_0     = (const float*)d_in[6];
  const float* W_l1_1     = (const float*)d_in[7];
  const float* W_fc0      = (const float*)d_in[8];
  const float* W_fc1      = (const float*)d_in[9];
  const float* W_l2_0     = (const float*)d_in[10];
  const float* W_l2_1     = (const float*)d_in[11];
  const int*   edge_index = (const int*)d_in[12];

  float* ws   = (float*)d_ws;
  float* y0   = ws;                          // [N,32]
  float* y1   = y0 + (size_t)N_NODES * 32;   // [N,48]
  float* s0   = y1 + (size_t)N_NODES * 48;   // [N,32]
  float* s1   = s0 + (size_t)N_NODES * 32;   // [N,48]
  float* agg0 = s1 + (size_t)N_NODES * 48;   // [N,48]
  float* agg1 = agg0 + (size_t)N_NODES * 48; // [N,144]

  // zero the scatter targets (agg0 followed immediately by agg1)
  hipMemsetAsync(agg0, 0, (size_t)N_NODES * (48 + 144) * sizeof(float), stream);

  int threads = 256;
  int gridNode = (N_NODES * 80 + threads - 1) / threads;
  tfn_node_prep<<<gridNode, threads, 0, stream>>>(
      node_input, node_attr, W_sc0, W_sc1, W_l1_0, W_l1_1, y0, y1, s0, s1);

  tfn_edge_kernel<<<1024, 128, 0, stream>>>(
      edge_scal, edge_attr, edge_index, W_fc0, W_fc1, y0, y1, agg0, agg1);

  tfn_node_out<<<gridNode, threads, 0, stream>>>(
      agg0, agg1, W_l2_0, W_l2_1, s0, s1, (float*)d_out);
}